// GeneratorCell_70617852281103
// MI455X (gfx1250) — compile-verified
//
#include <hip/hip_runtime.h>

typedef __attribute__((ext_vector_type(2))) float v2f;
typedef __attribute__((ext_vector_type(8))) float v8f;

#define Bsz 128
#define Mq  32
#define Nn  64
#define Kk  2048
#define PLEN (Nn + Kk - 1) /* 2111 */

// ---------------------------------------------------------------------------
// Kernel 1: conv(M+1,1) + GRU gate + linear -> coef[b,n] = it * alpha
// One block per batch, one thread per n (64 threads = 2 wave32).
// Also emits ht to its slot in d_out.
// ---------------------------------------------------------------------------
__global__ __launch_bounds__(Nn) void gen_cell_state_kernel(
    const float* __restrict__ qt1,   const float* __restrict__ ht1,
    const float* __restrict__ zt,    const float* __restrict__ alpha,
    const float* __restrict__ conv_w,const float* __restrict__ conv_b,
    const float* __restrict__ W_xr,  const float* __restrict__ W_hr, const float* __restrict__ b_r,
    const float* __restrict__ W_xu,  const float* __restrict__ W_hu, const float* __restrict__ b_u,
    const float* __restrict__ W_xn,  const float* __restrict__ W_hn, const float* __restrict__ b_n,
    const float* __restrict__ lin_w, const float* __restrict__ lin_b,
    float* __restrict__ out_ht, float* __restrict__ coef)
{
    __shared__ float sh_ht[2 * Nn];
    const int b = blockIdx.x;
    const int n = threadIdx.x;

    // Ht1[b,n,o] = sum_{h,c} Mt1[b,h,n,c] * conv_w[o,c,h] + conv_b[o]
    // conv_w flat (o,c,h): o*66 + c*33 + h
    float h0 = conv_b[0], h1 = conv_b[1];
    const float* q = qt1 + (((size_t)b * Mq) * Nn + n) * 2;
    #pragma unroll 4
    for (int h = 0; h < Mq; ++h) {
        float m0 = q[(size_t)h * Nn * 2 + 0];
        float m1 = q[(size_t)h * Nn * 2 + 1];
        h0 = fmaf(m0, conv_w[h],      fmaf(m1, conv_w[33 + h], h0));
        h1 = fmaf(m0, conv_w[66 + h], fmaf(m1, conv_w[99 + h], h1));
    }
    const float hp0 = ht1[((size_t)b * Nn + n) * 2 + 0];
    const float hp1 = ht1[((size_t)b * Nn + n) * 2 + 1];
    h0 = fmaf(hp0, conv_w[32], fmaf(hp1, conv_w[65],  h0));
    h1 = fmaf(hp0, conv_w[98], fmaf(hp1, conv_w[131], h1));

    const float x0 = zt[(size_t)b * Nn + n];
    const float x1 = alpha[(size_t)b * Nn + n];

    auto sig = [](float v) { return 1.0f / (1.0f + expf(-v)); };
    // xt @ W + Ht1 @ Wh + bias ; W[i,o] flat = W[i*2+o]
    float r0 = sig(x0*W_xr[0] + x1*W_xr[2] + h0*W_hr[0] + h1*W_hr[2] + b_r[0]);
    float r1 = sig(x0*W_xr[1] + x1*W_xr[3] + h0*W_hr[1] + h1*W_hr[3] + b_r[1]);
    float u0 = sig(x0*W_xu[0] + x1*W_xu[2] + h0*W_hu[0] + h1*W_hu[2] + b_u[0]);
    float u1 = sig(x0*W_xu[1] + x1*W_xu[3] + h0*W_hu[1] + h1*W_hu[3] + b_u[1]);
    float g0 = r0 * h0, g1 = r1 * h1;
    float t0 = tanhf(x0*W_xn[0] + x1*W_xn[2] + g0*W_hn[0] + g1*W_hn[2] + b_n[0]);
    float t1 = tanhf(x0*W_xn[1] + x1*W_xn[3] + g0*W_hn[1] + g1*W_hn[3] + b_n[1]);
    float htv0 = u0 * hp0 + (1.0f - u0) * t0;
    float htv1 = u1 * hp1 + (1.0f - u1) * t1;

    sh_ht[2 * n + 0] = htv0;
    sh_ht[2 * n + 1] = htv1;
    out_ht[((size_t)b * Nn + n) * 2 + 0] = htv0;
    out_ht[((size_t)b * Nn + n) * 2 + 1] = htv1;
    __syncthreads();

    // it[b,n] = sum_j ht_flat[b,j] * lin_w[n,j] + lin_b[n] ; coef = it * alpha
    float acc = lin_b[n];
    const float* lw = lin_w + (size_t)n * 2 * Nn;
    #pragma unroll 8
    for (int j = 0; j < 2 * Nn; ++j) acc = fmaf(sh_ht[j], lw[j], acc);
    coef[(size_t)b * Nn + n] = acc * x1;
}

// ---------------------------------------------------------------------------
// Kernel 2: f_srt[b,p,c] = sum_{i=0..63} coef[b,i]*W_emb[p-i,c] via f32 WMMA,
// fused with the st / queue-shift scatter. One wave32 per 16x16 output tile.
// The Toeplitz B operand is staged as an 80-float zero-padded W_emb window in
// LDS stored in REVERSED k order, so each B fragment is an adjacent ascending
// LDS pair in fragment order (no swap movs): the fully unrolled K-loop is
// ds_load(+2addr) + global b64 + v_wmma_f32_16x16x4_f32 only, branch-free.
// Grid: (B/16, ceil(2112/16)=132 position tiles, 2 channels).
// ---------------------------------------------------------------------------
__global__ __launch_bounds__(32) void gen_cell_wmma_overlap_add(
    const float* __restrict__ coef, const float* __restrict__ W_emb,
    const float* __restrict__ qt1,
    float* __restrict__ out_st, float* __restrict__ out_qt)
{
    __shared__ float semb[80];   // semb[t] = W_emb[p0+15-t, c], zero-padded

    const int lane  = threadIdx.x;
    const int m16   = lane & 15;   // A: row (batch) | B,D: column (position)
    const int khalf = lane >> 4;
    const int b0 = blockIdx.x * 16;
    const int p0 = blockIdx.y * 16;
    const int c  = blockIdx.z;

    // Stage reversed waveform window: indices p-i span [p0-63, p0+15].
    const int ktop = p0 + 15;
    #pragma unroll
    for (int t = lane; t < 80; t += 32) {
        const int k = ktop - t;
        semb[t] = ((unsigned)k < (unsigned)Kk) ? W_emb[(size_t)k * 2 + c] : 0.0f;
    }
    __syncthreads();

    const float* crow = coef + (size_t)(b0 + m16) * Nn;
    const float* sbc  = semb + (15 - m16);  // sbc[i] = W_emb[p0+m16-i, c]
    const int p = p0 + m16;

    v8f acc = {0.f, 0.f, 0.f, 0.f, 0.f, 0.f, 0.f, 0.f};

    // K=64 reduction in 16 steps of V_WMMA_F32_16X16X4_F32.
    // A (16x4 f32): vgpr j, lane l -> A[l&15, j + 2*(l>>4)]
    // B (4x16 f32): vgpr j, lane l -> B[j + 2*(l>>4), l&15], B[i,n] = W_emb[p0+n - i, c]
    #pragma unroll
    for (int ic = 0; ic < 16; ++ic) {
        const int i0 = 4 * ic + 2 * khalf;
        v2f A, Bv;
        A[0]  = crow[i0];
        A[1]  = crow[i0 + 1];
        Bv[0] = sbc[i0];       // W_emb[p - i0, c]      (zero-padded)
        Bv[1] = sbc[i0 + 1];   // W_emb[p - i0 - 1, c]  (zero-padded)
        acc = __builtin_amdgcn_wmma_f32_16x16x4_f32(
            /*neg_a=*/false, A, /*neg_b=*/false, Bv,
            /*c_mod=*/(short)0, acc, /*reuse_a=*/false, /*reuse_b=*/false);
    }

    // D layout: vgpr v, lane l -> D[v + 8*(l>>4), l&15]  (row=batch, col=position)
    #pragma unroll
    for (int v = 0; v < 8; ++v) {
        const int b = b0 + v + 8 * khalf;
        const float f = acc[v];
        if (p < Nn) {
            // st = qt1[:,0] + f_srt[:, :N]
            out_st[((size_t)b * Nn + p) * 2 + c] =
                qt1[(((size_t)b * Mq + 0) * Nn + p) * 2 + c] + f;
        } else {
            // qt flat index j = p - N in [0, 2048); tail valid for j < 2047
            const int j = p - Nn;
            const int mrow = j >> 6, nn = j & 63;
            const float qs = (mrow < Mq - 1)
                ? qt1[(((size_t)b * Mq + mrow + 1) * Nn + nn) * 2 + c] : 0.0f;
            const float tail = (j < Mq * Nn - 1) ? f : 0.0f;
            out_qt[((size_t)b * Mq * Nn + j) * 2 + c] = tail + qs;
        }
    }
}

// ---------------------------------------------------------------------------
extern "C" void kernel_launch(void* const* d_in, const int* in_sizes, int n_in,
                              void* d_out, int out_size, void* d_ws, size_t ws_size,
                              hipStream_t stream) {
    (void)in_sizes; (void)n_in; (void)out_size; (void)ws_size;

    const float* qt1    = (const float*)d_in[0];
    const float* ht1    = (const float*)d_in[1];
    const float* zt     = (const float*)d_in[2];
    const float* alpha  = (const float*)d_in[3];
    const float* conv_w = (const float*)d_in[4];
    const float* conv_b = (const float*)d_in[5];
    const float* W_xr   = (const float*)d_in[6];
    const float* W_hr   = (const float*)d_in[7];
    const float* b_r    = (const float*)d_in[8];
    const float* W_xu   = (const float*)d_in[9];
    const float* W_hu   = (const float*)d_in[10];
    const float* b_u    = (const float*)d_in[11];
    const float* W_xn   = (const float*)d_in[12];
    const float* W_hn   = (const float*)d_in[13];
    const float* b_n    = (const float*)d_in[14];
    const float* lin_w  = (const float*)d_in[15];
    const float* lin_b  = (const float*)d_in[16];
    const float* W_emb  = (const float*)d_in[17];

    float* out    = (float*)d_out;
    float* out_st = out;                                   // [B,N,2]
    float* out_qt = out + (size_t)Bsz * Nn * 2;            // [B,M,N,2]
    float* out_ht = out_qt + (size_t)Bsz * Mq * Nn * 2;    // [B,N,2]

    float* coef = (float*)d_ws;                            // [B,N] f32 (32 KB)

    gen_cell_state_kernel<<<dim3(Bsz), dim3(Nn), 0, stream>>>(
        qt1, ht1, zt, alpha, conv_w, conv_b,
        W_xr, W_hr, b_r, W_xu, W_hu, b_u, W_xn, W_hn, b_n,
        lin_w, lin_b, out_ht, coef);

    gen_cell_wmma_overlap_add<<<dim3(Bsz / 16, (PLEN + 16) / 16, 2), dim3(32), 0, stream>>>(
        coef, W_emb, qt1, out_st, out_qt);
}